// CRF_75204877353436
// MI455X (gfx1250) — compile-verified
//
#include <hip/hip_runtime.h>
#include <math.h>
#include <stdint.h>

typedef float v2f __attribute__((ext_vector_type(2)));
typedef float v8f __attribute__((ext_vector_type(8)));

#define BSZ 256
#define TSZ 512
#define KSZ 128
#define MT  16     // batch tile = WMMA M
#define LDA 132    // padded LDS row stride (132 % 64 != 0 -> conflict-free column reads)

// ---------------------------------------------------------------------------
// Forward kernel: per-WG 16-batch tile, full T recursion.
// alpha_next = em[t] + rowmax(alpha) + log( exp(alpha - rowmax) @ exp(trans) )
// GEMM done with V_WMMA_F32_16X16X4_F32 (exact fp32).
// exp/log use the 1-ulp hardware transcendentals (v_exp_f32 / v_log_f32,
// TRANS32 ops that co-execute with VALU) since they sit on the serial
// critical path 511 times.
// ---------------------------------------------------------------------------
__global__ __launch_bounds__(256)
void crf_forward_kernel(const float* __restrict__ em,      // [B][T][K]
                        const float* __restrict__ trans,   // [K][K]
                        const float* __restrict__ start,   // [K]
                        const float* __restrict__ endv,    // [K]
                        const uint8_t* __restrict__ mask,  // [B][T] (bool)
                        float* __restrict__ logZ)          // [B]
{
  __shared__ __align__(16) float alphaT[MT * LDA];
  __shared__ __align__(16) float pT[MT * LDA];
  __shared__ float rowMax[MT];

  const int tid  = threadIdx.x;
  const int lane = tid & 31;
  const int wave = tid >> 5;      // 0..7 -> 16-wide N slice
  const int half = lane >> 4;     // 0 or 1 (half-wave)
  const int l    = lane & 15;
  const int n0   = wave * 16;
  const int b0   = blockIdx.x * MT;

  // phase-a element mapping: each thread owns row (tid>>4), 8 consecutive cols
  const int arow = tid >> 4;
  const int ac0  = (tid & 15) * 8;

  // ---- preload B fragments: E = exp(trans) for this wave's N slice --------
  // B is 4x16 (KxN). Lanes 0-15: v0=K0,v1=K1 ; lanes 16-31: v0=K2,v1=K3.
  v2f bfrag[32];
#pragma unroll
  for (int kk = 0; kk < 32; ++kk) {
    const int k0 = kk * 4 + half * 2;
    v2f b;
    b.x = __expf(trans[(k0 + 0) * KSZ + n0 + l]);
    b.y = __expf(trans[(k0 + 1) * KSZ + n0 + l]);
    bfrag[kk] = b;
  }

  // ---- alpha0 = start + emissions[:,0,:] ----------------------------------
  {
    const size_t ebase = (size_t)(b0 + arow) * TSZ * KSZ;
#pragma unroll
    for (int i = 0; i < 8; ++i) {
      const int c = ac0 + i;
      alphaT[arow * LDA + c] = start[c] + em[ebase + c];
    }
  }
  __syncthreads();

  // ---- recursion ----------------------------------------------------------
  for (int t = 1; t < TSZ; ++t) {
    // phase (a): row max (shfl reduce within 16-lane group == one row) + exp
    float av[8];
    float mx = -__builtin_inff();
#pragma unroll
    for (int i = 0; i < 8; ++i) {
      av[i] = alphaT[arow * LDA + ac0 + i];
      mx = fmaxf(mx, av[i]);
    }
#pragma unroll
    for (int off = 1; off < 16; off <<= 1)
      mx = fmaxf(mx, __shfl_xor(mx, off, 16));
    if ((tid & 15) == 0) rowMax[arow] = mx;
#pragma unroll
    for (int i = 0; i < 8; ++i)
      pT[arow * LDA + ac0 + i] = __expf(av[i] - mx);
    __syncthreads();

    // phase (b): S = P @ E for this wave's 16 columns; 4 independent
    // accumulators to break the WMMA dependency chain.
    v8f a0 = {}, a1 = {}, a2 = {}, a3 = {};
#pragma unroll
    for (int kk = 0; kk < 32; kk += 4) {
      // A 16x4: lane l reads P[l][4kk + 2*half + {0,1}]  (8B-aligned LDS b64)
      v2f fa0 = *(const v2f*)&pT[l * LDA + (kk + 0) * 4 + half * 2];
      v2f fa1 = *(const v2f*)&pT[l * LDA + (kk + 1) * 4 + half * 2];
      v2f fa2 = *(const v2f*)&pT[l * LDA + (kk + 2) * 4 + half * 2];
      v2f fa3 = *(const v2f*)&pT[l * LDA + (kk + 3) * 4 + half * 2];
      a0 = __builtin_amdgcn_wmma_f32_16x16x4_f32(false, fa0, false, bfrag[kk + 0], (short)0, a0, false, false);
      a1 = __builtin_amdgcn_wmma_f32_16x16x4_f32(false, fa1, false, bfrag[kk + 1], (short)0, a1, false, false);
      a2 = __builtin_amdgcn_wmma_f32_16x16x4_f32(false, fa2, false, bfrag[kk + 2], (short)0, a2, false, false);
      a3 = __builtin_amdgcn_wmma_f32_16x16x4_f32(false, fa3, false, bfrag[kk + 3], (short)0, a3, false, false);
    }
    v8f acc = (a0 + a1) + (a2 + a3);

    // phase (c): alpha_new = em + rowmax + log(S), masked write-back.
    // D layout: lanes 0-15: VGPR r -> M=r ; lanes 16-31: M=r+8 ; N = l.
    const int j = n0 + l;
#pragma unroll
    for (int r = 0; r < 8; ++r) {
      const int m = r + half * 8;
      const int b = b0 + m;
      const float old = alphaT[m * LDA + j];
      const float e   = em[(size_t)b * TSZ * KSZ + (size_t)t * KSZ + j];
      const float v   = e + rowMax[m] + __logf(acc[r]);
      alphaT[m * LDA + j] = mask[b * TSZ + t] ? v : old;
      if (t + 1 < TSZ)  // warm L2/L0 for next step's emissions tile
        __builtin_prefetch(&em[(size_t)b * TSZ * KSZ + (size_t)(t + 1) * KSZ + j], 0, 1);
    }
    __syncthreads();
  }

  // ---- logZ = logsumexp(alpha + end) per row ------------------------------
  {
    float av[8];
    float mx = -__builtin_inff();
#pragma unroll
    for (int i = 0; i < 8; ++i) {
      const int c = ac0 + i;
      av[i] = alphaT[arow * LDA + c] + endv[c];
      mx = fmaxf(mx, av[i]);
    }
#pragma unroll
    for (int off = 1; off < 16; off <<= 1)
      mx = fmaxf(mx, __shfl_xor(mx, off, 16));
    float s = 0.f;
#pragma unroll
    for (int i = 0; i < 8; ++i) s += __expf(av[i] - mx);
#pragma unroll
    for (int off = 1; off < 16; off <<= 1)
      s += __shfl_xor(s, off, 16);
    if ((tid & 15) == 0) logZ[b0 + arow] = mx + __logf(s);
  }
}

// ---------------------------------------------------------------------------
// Score + final reduction: one block, thread b handles one sequence.
// ---------------------------------------------------------------------------
__global__ __launch_bounds__(256)
void crf_score_kernel(const float* __restrict__ em,
                      const float* __restrict__ trans,
                      const float* __restrict__ start,
                      const float* __restrict__ endv,
                      const int* __restrict__ tags,
                      const uint8_t* __restrict__ mask,
                      const float* __restrict__ logZ,
                      float* __restrict__ out)
{
  __shared__ float red[256];
  const int b = threadIdx.x;
  const int* tg = tags + b * TSZ;
  const uint8_t* mk = mask + b * TSZ;
  const float* e = em + (size_t)b * TSZ * KSZ;

  int prev = tg[0];
  float sc = start[prev] + e[prev];
  int cnt = mk[0] ? 1 : 0;
  for (int t = 1; t < TSZ; ++t) {
    const int cur = tg[t];
    const float m = mk[t] ? 1.0f : 0.0f;
    sc += (e[(size_t)t * KSZ + cur] + trans[prev * KSZ + cur]) * m;
    cnt += mk[t] ? 1 : 0;
    prev = cur;
  }
  sc += endv[tg[cnt - 1]];

  red[b] = logZ[b] - sc;
  __syncthreads();
#pragma unroll
  for (int s = 128; s > 0; s >>= 1) {
    if (b < s) red[b] += red[b + s];
    __syncthreads();
  }
  if (b == 0) out[0] = red[0] * (1.0f / BSZ);
}

// ---------------------------------------------------------------------------
extern "C" void kernel_launch(void* const* d_in, const int* in_sizes, int n_in,
                              void* d_out, int out_size, void* d_ws, size_t ws_size,
                              hipStream_t stream) {
  const float*   em    = (const float*)d_in[0];    // emissions [B][T][K]
  const float*   trans = (const float*)d_in[1];    // [K][K]
  const float*   start = (const float*)d_in[2];    // [K]
  const float*   endv  = (const float*)d_in[3];    // [K]
  const int*     tags  = (const int*)d_in[4];      // [B][T]
  const uint8_t* mask  = (const uint8_t*)d_in[5];  // [B][T] bool
  float* logZ = (float*)d_ws;                      // [B] scratch

  crf_forward_kernel<<<BSZ / MT, 256, 0, stream>>>(em, trans, start, endv, mask, logZ);
  crf_score_kernel<<<1, 256, 0, stream>>>(em, trans, start, endv, tags, mask, logZ, (float*)d_out);
}